// SparseNN_20890720927986
// MI455X (gfx1250) — compile-verified
//
#include <hip/hip_runtime.h>

typedef __attribute__((ext_vector_type(2))) float v2f;
typedef __attribute__((ext_vector_type(8))) float v8f;

#define NUM_FEATURES 135909
#define HLS          128
#define NUM_LABELS   670091
#define BATCH        256
#define CHUNK_K      8192
#define NCHUNK       ((NUM_FEATURES + CHUNK_K - 1) / CHUNK_K)   // 17

// ---------------------------------------------------------------------------
// Layer 1, phase A: split-K partial dot products, sparsity-aware.
// One block = (row m, K-chunk c); thread t owns output feature n = t.
// All 128 threads broadcast-load the same x[m,k] (single cacheline per wave),
// and only touch W1 / issue FMAs when a 4-group of x contains a nonzero
// (~0.4% of iterations at 0.1% density). Fixed iteration order per thread
// -> bitwise deterministic. No atomics: partials reduced by l1_finalize.
// ---------------------------------------------------------------------------
__global__ void l1_partial(const float* __restrict__ x,
                           const float* __restrict__ w1,
                           float* __restrict__ partial) {
  const int n = threadIdx.x;          // 0..127: output feature
  const int m = blockIdx.x;           // batch row
  const int c = blockIdx.y;           // K chunk
  const int k0   = c * CHUNK_K;
  const int kend = (k0 + CHUNK_K < NUM_FEATURES) ? (k0 + CHUNK_K) : NUM_FEATURES;

  const float* __restrict__ xrow = x  + (size_t)m * NUM_FEATURES;
  const float* __restrict__ wrow = w1 + (size_t)n * NUM_FEATURES;

  float acc = 0.0f;
  int k = k0;
  for (; k + 4 <= kend; k += 4) {
    const float x0 = xrow[k + 0];
    const float x1 = xrow[k + 1];
    const float x2 = xrow[k + 2];
    const float x3 = xrow[k + 3];
    // predicate is identical across the whole block -> wave-uniform branch
    if ((x0 != 0.0f) | (x1 != 0.0f) | (x2 != 0.0f) | (x3 != 0.0f)) {
      acc = fmaf(x0, wrow[k + 0], acc);
      acc = fmaf(x1, wrow[k + 1], acc);
      acc = fmaf(x2, wrow[k + 2], acc);
      acc = fmaf(x3, wrow[k + 3], acc);
    }
  }
  for (; k < kend; ++k) {             // K % 4 tail (135909 % 4 == 1)
    const float xv = xrow[k];
    if (xv != 0.0f) acc = fmaf(xv, wrow[k], acc);
  }
  partial[((size_t)c * BATCH + m) * HLS + n] = acc;
}

// ---------------------------------------------------------------------------
// Layer 1, phase B: fixed-order chunk reduction + bias + ReLU -> h[256x128].
// ---------------------------------------------------------------------------
__global__ void l1_finalize(const float* __restrict__ partial,
                            const float* __restrict__ b1,
                            float* __restrict__ h) {
  const int n = threadIdx.x;          // 0..127
  const int m = blockIdx.x;           // 0..255
  float s = b1[n];
#pragma unroll
  for (int c = 0; c < NCHUNK; ++c)
    s += partial[((size_t)c * BATCH + m) * HLS + n];
  h[(size_t)m * HLS + n] = s > 0.0f ? s : 0.0f;
}

// ---------------------------------------------------------------------------
// Layer 2: out[256 x 670091] = h @ W2^T + b2 via V_WMMA_F32_16X16X4_F32.
// Block = 256 threads = 8 waves. Each wave owns one 16-label N-tile:
//   - preloads the full 128(K) x 16(N) B fragment of W2^T into 64 VGPRs,
//   - sweeps all 16 M-tiles (BATCH=256), 32 k-steps each, reusing B,
//   - initializes the accumulator with the per-column bias,
//   - stores the 16x16 f32 tile (predicated on n < NUM_LABELS for the tail).
// h (128 KB) is L2-resident; A-fragment loads are 8B-aligned global_load_b64.
// EXEC stays all-1s around every WMMA (tail handled by address clamping).
// ---------------------------------------------------------------------------
__global__ void __launch_bounds__(256, 1)
l2_wmma(const float* __restrict__ h,
        const float* __restrict__ w2,
        const float* __restrict__ b2,
        float* __restrict__ out) {
  const int lane   = threadIdx.x & 31;
  const int wave   = threadIdx.x >> 5;
  const int laneN  = lane & 15;        // column within 16-wide tile
  const int laneHi = lane >> 4;        // 0: K pair {0,1}; 1: K pair {2,3}

  const long NT    = (NUM_LABELS + 15) / 16;               // 41881 N-tiles
  const long ntile = (long)blockIdx.x * 8 + wave;
  if (ntile >= NT) return;                                  // wave-uniform exit

  const int n  = (int)ntile * 16 + laneN;
  const int nc = (n < NUM_LABELS) ? n : (NUM_LABELS - 1);   // clamp for loads

  // Preload B fragment: w2 is [NUM_LABELS, 128] row-major, B(k,n) = w2[n,k].
  const float* __restrict__ wrow = w2 + (size_t)nc * HLS + 2 * laneHi;
  v2f B[32];
#pragma unroll
  for (int ks = 0; ks < 32; ++ks)
    B[ks] = *(const v2f*)(wrow + 4 * ks);

  const float bias = b2[nc];

#pragma unroll 1
  for (int mt = 0; mt < 16; ++mt) {
    const int m0 = mt * 16;
    const float* __restrict__ hrow =
        h + (size_t)(m0 + laneN) * HLS + 2 * laneHi;

    v8f acc;
#pragma unroll
    for (int r = 0; r < 8; ++r) acc[r] = bias;   // bias depends only on N

#pragma unroll
    for (int ks = 0; ks < 32; ++ks) {
      v2f a = *(const v2f*)(hrow + 4 * ks);
      acc = __builtin_amdgcn_wmma_f32_16x16x4_f32(
          /*neg_a=*/false, a, /*neg_b=*/false, B[ks],
          /*c_mod=*/(short)0, acc, /*reuse_a=*/false, /*reuse_b=*/false);
    }

    if (n < NUM_LABELS) {
#pragma unroll
      for (int r = 0; r < 8; ++r) {
        const int mm = m0 + laneHi * 8 + r;      // C layout: VGPR r -> M row
        out[(size_t)mm * NUM_LABELS + n] = acc[r];
      }
    }
  }
}

// ---------------------------------------------------------------------------
// Host launch. Workspace layout (all rewritten every call, no stale state):
//   [0, 128KB)            h        : 256 x 128 f32
//   [128KB, 128KB+2.2MB)  partial  : NCHUNK x 256 x 128 f32
// ---------------------------------------------------------------------------
extern "C" void kernel_launch(void* const* d_in, const int* in_sizes, int n_in,
                              void* d_out, int out_size, void* d_ws, size_t ws_size,
                              hipStream_t stream) {
  const float* x  = (const float*)d_in[0];
  const float* w1 = (const float*)d_in[1];
  const float* b1 = (const float*)d_in[2];
  const float* w2 = (const float*)d_in[3];
  const float* b2 = (const float*)d_in[4];
  float* out = (float*)d_out;

  float* h       = (float*)d_ws;
  float* partial = h + (size_t)BATCH * HLS;

  l1_partial<<<dim3(BATCH, NCHUNK), 128, 0, stream>>>(x, w1, partial);
  l1_finalize<<<BATCH, HLS, 0, stream>>>(partial, b1, h);

  const long NT = (NUM_LABELS + 15) / 16;       // 41881 N-tiles
  const int  nblocks = (int)((NT + 7) / 8);     // 8 N-tiles (waves) per block
  l2_wmma<<<nblocks, 256, 0, stream>>>(h, w2, b2, out);
}